// Yolo_Loss_84344567759441
// MI455X (gfx1250) — compile-verified
//
#include <hip/hip_runtime.h>
#include <math.h>

typedef __attribute__((ext_vector_type(2))) float v2f;
typedef __attribute__((ext_vector_type(8))) float v8f;

#define S_N    13
#define A_N    5
#define C_N    20
#define N_GT   32
#define CELLS  (S_N * S_N)        // 169
#define ACELLS (CELLS * A_N)      // 845
#define CH     (5 + C_N)          // 25

__constant__ float d_AX[A_N] = {1.3221f, 3.19275f, 5.05587f, 9.47112f, 11.2364f};
__constant__ float d_AY[A_N] = {1.73145f, 4.00944f, 8.09892f, 4.84053f, 10.0071f};

// One block per batch element. 256 threads = 8 wave32s.
__global__ __launch_bounds__(256) void yolo_main(
    const float* __restrict__ pred,   // (B,13,13,125)
    const float* __restrict__ gtb,    // (B,32,4) corner, [0,1]
    const int*   __restrict__ gtl,    // (B,32) labels 1..20
    float*       __restrict__ partial)// (B,5) per-block loss partials
{
    const int b   = blockIdx.x;
    const int tid = threadIdx.x;

    __shared__ float sGt[N_GT][4];   // gt corners at scale 13
    __shared__ int   sKey[N_GT];     // (cy*13+cx)*5 + j  == flat anchor-cell idx
    __shared__ float sT[N_GT][4];    // tx, ty, tw, th
    __shared__ int   sLbl[N_GT];     // 0-based class label
    __shared__ float sRed[8][5];

    // --- Stage 1: GT -> responsible anchor matching (32 threads) ---
    if (tid < N_GT) {
        const float* g = gtb + ((size_t)b * N_GT + tid) * 4;
        float x0 = g[0] * (float)S_N, y0 = g[1] * (float)S_N;
        float x1 = g[2] * (float)S_N, y1 = g[3] * (float)S_N;
        sGt[tid][0] = x0; sGt[tid][1] = y0; sGt[tid][2] = x1; sGt[tid][3] = y1;

        float bx = 0.5f * (x0 + x1), by = 0.5f * (y0 + y1);
        float w  = x1 - x0,          h  = y1 - y0;
        float fx = floorf(bx), fy = floorf(by);
        int cx = (int)fx; cx = min(max(cx, 0), S_N - 1);
        int cy = (int)fy; cy = min(max(cy, 0), S_N - 1);
        float ccx = (float)cx + 0.5f, ccy = (float)cy + 0.5f;

        float gArea = w * h;
        int   j = 0; float best = -1.0f;
        #pragma unroll
        for (int a = 0; a < A_N; ++a) {
            float aw = d_AX[a], ah = d_AY[a];
            float ax0 = ccx - 0.5f * aw, ay0 = ccy - 0.5f * ah;
            float ax1 = ccx + 0.5f * aw, ay1 = ccy + 0.5f * ah;
            float lx = fmaxf(ax0, x0), ly = fmaxf(ay0, y0);
            float rx = fminf(ax1, x1), ry = fminf(ay1, y1);
            float iw = fmaxf(rx - lx, 0.0f), ih = fmaxf(ry - ly, 0.0f);
            float inter = iw * ih;
            float iou = inter / (aw * ah + gArea - inter);
            if (iou > best) { best = iou; j = a; }
        }
        sKey[tid]  = (cy * S_N + cx) * A_N + j;
        sT[tid][0] = bx - fx;            // txty uses un-clamped floor (matches ref)
        sT[tid][1] = by - fy;
        sT[tid][2] = w / d_AX[j];
        sT[tid][3] = h / d_AY[j];
        sLbl[tid]  = gtl[(size_t)b * N_GT + tid] - 1;
    }
    __syncthreads();

    // --- Stage 2: stream 845 anchor-cells, accumulate 5 loss terms ---
    float accXY = 0.f, accWH = 0.f, accC = 0.f, accNC = 0.f, accCls = 0.f;

    for (int idx = tid; idx < ACELLS; idx += blockDim.x) {
        int s  = idx / A_N;
        int a  = idx - s * A_N;
        int cy = s / S_N;
        int cx = s - cy * S_N;
        const float* p = pred + ((size_t)b * CELLS + s) * (CELLS == 169 ? 125 : 125) + a * CH;
        // prefetch next iteration's chunk (256 anchor-cells * 25 floats ahead)
        __builtin_prefetch((const void*)(p + 256 * CH), 0, 1);

        float px = 1.0f / (1.0f + expf(-p[0]));
        float py = 1.0f / (1.0f + expf(-p[1]));
        float pw = expf(p[2]);
        float ph = expf(p[3]);
        float pc = 1.0f / (1.0f + expf(-p[4]));

        // log-softmax normalizer over 20 class logits
        float mx = -3.402823466e+38f;
        float cl[C_N];
        #pragma unroll
        for (int c = 0; c < C_N; ++c) { cl[c] = p[5 + c]; mx = fmaxf(mx, cl[c]); }
        float se = 0.f;
        #pragma unroll
        for (int c = 0; c < C_N; ++c) se += expf(cl[c] - mx);
        float lse = mx + logf(se);

        // predicted corner box (scale 13)
        float cpx = (float)cx + 0.5f + px;
        float cpy = (float)cy + 0.5f + py;
        float wp  = d_AX[a] * pw;
        float hp  = d_AY[a] * ph;
        float px0 = cpx - 0.5f * wp, py0 = cpy - 0.5f * hp;
        float px1 = cpx + 0.5f * wp, py1 = cpy + 0.5f * hp;
        float pArea = wp * hp;

        // gt_conf = max IoU vs 32 gt boxes (LDS)
        float gconf = 0.f;
        #pragma unroll 4
        for (int n = 0; n < N_GT; ++n) {
            float gx0 = sGt[n][0], gy0 = sGt[n][1], gx1 = sGt[n][2], gy1 = sGt[n][3];
            float lx = fmaxf(px0, gx0), ly = fmaxf(py0, gy0);
            float rx = fminf(px1, gx1), ry = fminf(py1, gy1);
            float iw = fmaxf(rx - lx, 0.0f), ih = fmaxf(ry - ly, 0.0f);
            float inter = iw * ih;
            float gArea = (gx1 - gx0) * (gy1 - gy0);
            float iou = inter / (pArea + gArea - inter);
            gconf = fmaxf(gconf, iou);
        }

        // responsible-anchor scan (last write wins, mimicking scatter order)
        int f = -1;
        #pragma unroll 8
        for (int n = 0; n < N_GT; ++n) if (sKey[n] == idx) f = n;

        float dC = gconf - pc;
        if (f >= 0) {
            float dx = sT[f][0] - px, dy = sT[f][1] - py;
            accXY += dx * dx + dy * dy;
            float dw = sqrtf(sT[f][2]) - sqrtf(pw);
            float dh = sqrtf(sT[f][3]) - sqrtf(ph);
            accWH += dw * dw + dh * dh;
            accC  += dC * dC;
            int lbl = sLbl[f];
            float logit = p[5 + lbl];     // re-read (cached), avoids dyn reg index
            accCls += lse - logit;        // -log_softmax[lbl]
        } else {
            accNC += dC * dC;
        }
    }

    // --- Stage 3: block reduce 5 accumulators ---
    float acc[5] = {accXY, accWH, accC, accNC, accCls};
    #pragma unroll
    for (int off = 16; off > 0; off >>= 1) {
        #pragma unroll
        for (int i = 0; i < 5; ++i) acc[i] += __shfl_down(acc[i], off, 32);
    }
    int lane = tid & 31, wid = tid >> 5;
    if (lane == 0) {
        #pragma unroll
        for (int i = 0; i < 5; ++i) sRed[wid][i] = acc[i];
    }
    __syncthreads();
    if (tid == 0) {
        #pragma unroll
        for (int i = 0; i < 5; ++i) {
            float r = 0.f;
            #pragma unroll
            for (int w = 0; w < 8; ++w) r += sRed[w][i];
            partial[(size_t)b * 5 + i] = r;
        }
    }
}

// Final reduction: 512x5 partials -> 6 outputs. One wave32, EXEC all-1s,
// cross-lane sums done with V_WMMA_F32_16X16X4_F32 (A=acc, B=ones):
// D[m][n] = acc[m] + acc[m+16]; per-lane sum of 8 C VGPRs + shfl_xor(16)
// yields the full 32-lane total.
__global__ __launch_bounds__(32) void yolo_reduce(
    const float* __restrict__ partial, float* __restrict__ out, int B)
{
    int lane = threadIdx.x;
    float s[5] = {0.f, 0.f, 0.f, 0.f, 0.f};
    for (int k = lane; k < B; k += 32) {
        const float* p = partial + (size_t)k * 5;
        #pragma unroll
        for (int i = 0; i < 5; ++i) s[i] += p[i];
    }

    v2f ones; ones[0] = 1.0f; ones[1] = 1.0f;
    float t[5];
    #pragma unroll
    for (int i = 0; i < 5; ++i) {
        v2f a; a[0] = s[i]; a[1] = 0.0f;
        v8f c = {};
        c = __builtin_amdgcn_wmma_f32_16x16x4_f32(
                /*neg_a=*/false, a, /*neg_b=*/false, ones,
                /*c_mod=*/(short)0, c, /*reuse_a=*/false, /*reuse_b=*/false);
        float sc = c[0] + c[1] + c[2] + c[3] + c[4] + c[5] + c[6] + c[7];
        sc += __shfl_xor(sc, 16, 32);   // combine row halves -> full 32-lane sum
        t[i] = sc;
    }

    if (lane == 0) {
        float l1 = 5.0f * t[0];
        float l2 = 5.0f * t[1];
        float l3 = 1.0f * t[2];
        float l4 = 0.5f * t[3];
        float l5 = 1.0f * t[4];
        out[0] = l1 + l2 + l3 + l4 + l5;
        out[1] = l1; out[2] = l2; out[3] = l3; out[4] = l4; out[5] = l5;
    }
}

extern "C" void kernel_launch(void* const* d_in, const int* in_sizes, int n_in,
                              void* d_out, int out_size, void* d_ws, size_t ws_size,
                              hipStream_t stream) {
    const float* pred = (const float*)d_in[0];
    const float* gtb  = (const float*)d_in[1];
    const int*   gtl  = (const int*)d_in[2];
    const int B = in_sizes[1] / (N_GT * 4);   // gt_boxes is (B,32,4)

    float* partials = (float*)d_ws;           // B*5 floats (10 KB for B=512)

    yolo_main<<<B, 256, 0, stream>>>(pred, gtb, gtl, partials);
    yolo_reduce<<<1, 32, 0, stream>>>(partials, (float*)d_out, B);
}